// GCLModel_77790447665862
// MI455X (gfx1250) — compile-verified
//
#include <hip/hip_runtime.h>

#define NUM_USERS 500000
#define NUM_ITEMS 500000
#define EMB 64
#define NUM_EDGES 1250000
#define NNODES (NUM_USERS + NUM_ITEMS)

#define WAVES 4          // waves per block in the GEMM kernel
#define LDSPITCH 80      // 64 + 16 padding: rows k and k+2 land in disjoint bank sets

typedef __attribute__((ext_vector_type(2))) float v2f;
typedef __attribute__((ext_vector_type(8))) float v8f;

// ---------------------------------------------------------------------------
// Stage 1: edge gather + scatter-add (mean aggregation numerator + degrees).
// One thread per (edge, component). A wave handles 32 contiguous components
// of a single edge -> coalesced 128B gather read, coalesced f32 atomics.
// ---------------------------------------------------------------------------
__global__ __launch_bounds__(256) void sage_aggregate(
    const int* __restrict__ src, const int* __restrict__ dst,
    const float* __restrict__ user, const float* __restrict__ item,
    float* __restrict__ agg, float* __restrict__ deg)
{
    unsigned long long t =
        (unsigned long long)blockIdx.x * blockDim.x + threadIdx.x;
    const unsigned long long total = (unsigned long long)NUM_EDGES * EMB;
    if (t >= total) return;

    const int e = (int)(t >> 6);     // edge id
    const int c = (int)(t & 63);     // embedding component

    const int s = src[e];
    const int d = dst[e];
    const float* xs = (s < NUM_USERS)
                        ? (user + (size_t)s * EMB)
                        : (item + (size_t)(s - NUM_USERS) * EMB);

    atomicAdd(agg + (size_t)d * EMB + c, xs[c]);   // global_atomic_add_f32
    if (c == 0) atomicAdd(deg + d, 1.0f);
}

// ---------------------------------------------------------------------------
// Stage 2: out = (agg/max(deg,1)) @ W_l + b_l + x @ W_r via f32 WMMA.
// Each wave owns 16 node rows and produces a 16x64 output slab as four
// 16x16 tiles, accumulating K=64 in 16 steps of V_WMMA_F32_16X16X4_F32 per
// GEMM (128 WMMAs per wave total).
//
// Operand layouts per CDNA5 ISA 7.12.2:
//   A 16x4 f32 : lanes 0-15 hold M=lane, K={4k,4k+1}; lanes 16-31 K={4k+2,4k+3}
//   B 4x16 f32 : lanes 0-15 hold N=lane, K={4k,4k+1}; lanes 16-31 K={4k+2,4k+3}
//   C/D 16x16  : VGPR r -> row r + 8*(lane>>4), col = lane&15
// ---------------------------------------------------------------------------
__global__ __launch_bounds__(WAVES * 32) void sage_gemm(
    const float* __restrict__ user, const float* __restrict__ item,
    const float* __restrict__ agg, const float* __restrict__ deg,
    const float* __restrict__ Wl, const float* __restrict__ bl,
    const float* __restrict__ Wr, float* __restrict__ out)
{
    __shared__ float sW[2 * 64 * LDSPITCH];   // 40 KB: W_l then W_r, padded rows

    for (int i = threadIdx.x; i < 64 * 64; i += WAVES * 32) {
        const int r = i >> 6, cc = i & 63;
        sW[r * LDSPITCH + cc]                 = Wl[i];
        sW[64 * LDSPITCH + r * LDSPITCH + cc] = Wr[i];
    }
    __syncthreads();

    const int lane = threadIdx.x & 31;
    const int wave = threadIdx.x >> 5;
    const int rit  = lane & 15;     // row index within A tile / col within B tile
    const int half = lane >> 4;     // selects K pair (A/B) and M half (C/D)
    const int node_base = (blockIdx.x * WAVES + wave) * 16;
    const int row  = node_base + rit;

    // Per-row mean scale: 1 / max(deg, 1)
    const float dval = deg[row];
    const float inv  = 1.0f / fmaxf(dval, 1.0f);

    const float* aggrow = agg + (size_t)row * EMB;
    const float* xrow = (row < NUM_USERS)
                          ? (user + (size_t)row * EMB)
                          : (item + (size_t)(row - NUM_USERS) * EMB);

    // Preload both A operand stacks (16 K-steps each) as 8-byte loads.
    v2f a_mean[16], a_x[16];
#pragma unroll
    for (int kb = 0; kb < 16; ++kb) {
        const int k0 = kb * 4 + half * 2;
        v2f am = *(const v2f*)(aggrow + k0);
        a_mean[kb] = am * inv;
        a_x[kb]    = *(const v2f*)(xrow + k0);
    }

#pragma unroll
    for (int nt = 0; nt < 4; ++nt) {
        const int col = nt * 16 + rit;
        v8f c = {};
#pragma unroll
        for (int kb = 0; kb < 16; ++kb) {        // mean @ W_l
            const int k0 = kb * 4 + half * 2;
            v2f b;
            b.x = sW[k0 * LDSPITCH + col];
            b.y = sW[(k0 + 1) * LDSPITCH + col];
            c = __builtin_amdgcn_wmma_f32_16x16x4_f32(
                    false, a_mean[kb], false, b, (short)0, c, false, false);
        }
#pragma unroll
        for (int kb = 0; kb < 16; ++kb) {        // x @ W_r
            const int k0 = kb * 4 + half * 2;
            v2f b;
            b.x = sW[64 * LDSPITCH + k0 * LDSPITCH + col];
            b.y = sW[64 * LDSPITCH + (k0 + 1) * LDSPITCH + col];
            c = __builtin_amdgcn_wmma_f32_16x16x4_f32(
                    false, a_x[kb], false, b, (short)0, c, false, false);
        }

        const float bias = bl[col];
#pragma unroll
        for (int r = 0; r < 8; ++r) {
            const int orow = node_base + r + 8 * half;
            out[(size_t)orow * EMB + col] = c[r] + bias;
        }
    }
}

extern "C" void kernel_launch(void* const* d_in, const int* in_sizes, int n_in,
                              void* d_out, int out_size, void* d_ws, size_t ws_size,
                              hipStream_t stream) {
    const int*   edge_index = (const int*)d_in[0];   // [2, E]
    const float* user       = (const float*)d_in[1]; // [NUM_USERS, 64]
    const float* item       = (const float*)d_in[2]; // [NUM_ITEMS, 64]
    const float* Wl         = (const float*)d_in[3]; // [64, 64]
    const float* bl         = (const float*)d_in[4]; // [64]
    const float* Wr         = (const float*)d_in[5]; // [64, 64]
    float*       out        = (float*)d_out;         // [N, 64] flat

    float* agg = (float*)d_ws;                        // [N, 64]
    float* deg = agg + (size_t)NNODES * EMB;          // [N]

    const size_t zero_bytes = ((size_t)NNODES * EMB + NNODES) * sizeof(float);
    hipMemsetAsync(d_ws, 0, zero_bytes, stream);

    const int* src = edge_index;
    const int* dst = edge_index + NUM_EDGES;

    {
        const unsigned long long total = (unsigned long long)NUM_EDGES * EMB;
        const int threads = 256;
        const unsigned int blocks = (unsigned int)((total + threads - 1) / threads);
        sage_aggregate<<<blocks, threads, 0, stream>>>(src, dst, user, item, agg, deg);
    }
    {
        const int threads = WAVES * 32;                  // 128
        const int blocks  = NNODES / (WAVES * 16);       // 1,000,000 / 64 = 15625 exact
        sage_gemm<<<blocks, threads, 0, stream>>>(user, item, agg, deg, Wl, bl, Wr, out);
    }
}